// Partitioner_20349555048507
// MI455X (gfx1250) — compile-verified
//
#include <hip/hip_runtime.h>
#include <hip/hip_bf16.h>
#include <math.h>

// ---------------------------------------------------------------------------
// GraphSAGE (2x SAGEConv mean-aggr) + MLP classifier + softmax, fp32.
// Matrix ops use V_WMMA_F32_16X16X4_F32 (exact fp32 on the CDNA5 matrix pipe).
// Workload is scatter/HBM bound (~2.8 GB -> ~120us @ 23.3 TB/s); GEMMs are
// only ~28 GFLOP so fp32 WMMA is the right precision (no reason to quantize).
// ---------------------------------------------------------------------------

typedef float v2f __attribute__((ext_vector_type(2)));
typedef float v8f __attribute__((ext_vector_type(8)));

#define N_NODES   50000        // 16 * 3125, exact 16-row tiling
#define N_EDGES   800000
#define IN_DIM    128
#define HIDDEN    256
#define NUM_SHARDS 64

__device__ __forceinline__ float gelu_exact(float x) {
    return 0.5f * x * (1.0f + erff(x * 0.7071067811865475f));
}

// ---- degree: one float atomic per edge ------------------------------------
__global__ void deg_kernel(const int* __restrict__ dst, float* __restrict__ deg) {
    int e = blockIdx.x * blockDim.x + threadIdx.x;
    if (e < N_EDGES) atomicAdd(&deg[dst[e]], 1.0f);
}

// ---- segment-sum scatter: block = edge, thread = feature (coalesced) ------
__global__ void scatter_kernel(const float* __restrict__ feat,
                               const int* __restrict__ src,
                               const int* __restrict__ dst,
                               float* __restrict__ agg, int F) {
    int e = blockIdx.x;
    int f = threadIdx.x;
    int s = src[e];
    int d = dst[e];
    atomicAdd(&agg[(size_t)d * F + f], feat[(size_t)s * F + f]);
}

// ---- fused dual-GEMM tile kernel:  out = act( A1*invdeg @ Wl + A2 @ Wr + b )
// One wave per 16x16 output tile, K-loop of WMMA f32 16x16x4.
template <int K, int H, int WPB>
__global__ void __launch_bounds__(WPB * 32)
fused_gemm(const float* __restrict__ A1, const float* __restrict__ Wl,
           const float* __restrict__ A2, const float* __restrict__ Wr,
           const float* __restrict__ bias, const float* __restrict__ deg,
           float* __restrict__ out, int doGelu)
{
    const int lane    = threadIdx.x & 31;
    const int wave    = threadIdx.x >> 5;
    const int m15     = lane & 15;        // M row within tile (A/B/C lane id)
    const int hi      = lane >> 4;        // selects K pair {0,1} vs {2,3}
    const int rowBase = blockIdx.x * 16;
    const int row     = rowBase + m15;
    const int tileN   = blockIdx.y * WPB + wave;
    const int col     = tileN * 16 + m15;

    float invd = 1.0f;
    if (deg) invd = 1.0f / fmaxf(deg[row], 1.0f);   // mean aggregation

    v8f c = {};
    for (int k0 = 0; k0 < K; k0 += 4) {
        const int ka = k0 + 2 * hi;
        // A: 16x4 f32, lane m15 = row, VGPR pair = K=ka,ka+1 (contiguous)
        v2f a = *(const v2f*)(A1 + (size_t)row * K + ka);
        a.x *= invd; a.y *= invd;
        // B: 4x16 f32, lane m15 = col, VGPR pair = K=ka,ka+1 (stride H)
        v2f b;
        b.x = Wl[(size_t)ka * H + col];
        b.y = Wl[(size_t)(ka + 1) * H + col];
        c = __builtin_amdgcn_wmma_f32_16x16x4_f32(false, a, false, b,
                                                  (short)0, c, false, false);
        if (A2) {  // root transform (uniform branch, EXEC unchanged)
            v2f a2 = *(const v2f*)(A2 + (size_t)row * K + ka);
            v2f b2;
            b2.x = Wr[(size_t)ka * H + col];
            b2.y = Wr[(size_t)(ka + 1) * H + col];
            c = __builtin_amdgcn_wmma_f32_16x16x4_f32(false, a2, false, b2,
                                                      (short)0, c, false, false);
        }
    }

    const float bb = bias[col];
#pragma unroll
    for (int r = 0; r < 8; ++r) {
        float v = c[r] + bb;
        if (doGelu) v = gelu_exact(v);
        out[(size_t)(rowBase + r + 8 * hi) * H + col] = v;
    }
}

// ---- final Linear(256,64) + softmax over 64 logits per row ----------------
__global__ void __launch_bounds__(128)
mlp2_softmax(const float* __restrict__ A, const float* __restrict__ W,
             const float* __restrict__ bias, float* __restrict__ out)
{
    __shared__ float lds[16 * NUM_SHARDS];
    const int lane    = threadIdx.x & 31;
    const int wave    = threadIdx.x >> 5;     // 4 waves -> 4 col tiles = 64 cols
    const int m15     = lane & 15;
    const int hi      = lane >> 4;
    const int rowBase = blockIdx.x * 16;
    const int row     = rowBase + m15;
    const int col     = wave * 16 + m15;

    v8f c = {};
    for (int k0 = 0; k0 < HIDDEN; k0 += 4) {
        const int ka = k0 + 2 * hi;
        v2f a = *(const v2f*)(A + (size_t)row * HIDDEN + ka);
        v2f b;
        b.x = W[(size_t)ka * NUM_SHARDS + col];
        b.y = W[(size_t)(ka + 1) * NUM_SHARDS + col];
        c = __builtin_amdgcn_wmma_f32_16x16x4_f32(false, a, false, b,
                                                  (short)0, c, false, false);
    }
    const float bb = bias[col];
#pragma unroll
    for (int r = 0; r < 8; ++r)
        lds[(r + 8 * hi) * NUM_SHARDS + col] = c[r] + bb;
    __syncthreads();

    if (threadIdx.x < 16) {
        const int r = threadIdx.x;
        const float* lr = &lds[r * NUM_SHARDS];
        float m = -INFINITY;
        for (int j = 0; j < NUM_SHARDS; ++j) m = fmaxf(m, lr[j]);
        float s = 0.0f;
        for (int j = 0; j < NUM_SHARDS; ++j) s += expf(lr[j] - m);
        const float inv = 1.0f / s;
        float* orow = out + (size_t)(rowBase + r) * NUM_SHARDS;
        for (int j = 0; j < NUM_SHARDS; ++j) orow[j] = expf(lr[j] - m) * inv;
    }
}

extern "C" void kernel_launch(void* const* d_in, const int* in_sizes, int n_in,
                              void* d_out, int out_size, void* d_ws, size_t ws_size,
                              hipStream_t stream)
{
    const float* x   = (const float*)d_in[0];
    const int*   ei  = (const int*)d_in[1];
    const int*   src = ei;            // edge_index[0]
    const int*   dst = ei + N_EDGES;  // edge_index[1]
    const float* Wl1 = (const float*)d_in[2];
    const float* bl1 = (const float*)d_in[3];
    const float* Wr1 = (const float*)d_in[4];
    const float* Wl2 = (const float*)d_in[5];
    const float* bl2 = (const float*)d_in[6];
    const float* Wr2 = (const float*)d_in[7];
    const float* Wc1 = (const float*)d_in[8];
    const float* bc1 = (const float*)d_in[9];
    const float* Wc2 = (const float*)d_in[10];
    const float* bc2 = (const float*)d_in[11];
    float* outp = (float*)d_out;

    // Workspace layout (floats): deg[N] | agg[N*256] | h1[N*256] | h2[N*256]
    float* ws  = (float*)d_ws;
    float* deg = ws;
    float* agg = deg + N_NODES;
    float* h1  = agg + (size_t)N_NODES * HIDDEN;
    float* h2  = h1  + (size_t)N_NODES * HIDDEN;

    const dim3 gGemm(N_NODES / 16, (HIDDEN / 16) / 8);   // (3125, 2), 8 waves/blk

    // degrees (shared by both layers)
    hipMemsetAsync(deg, 0, (size_t)N_NODES * sizeof(float), stream);
    deg_kernel<<<(N_EDGES + 255) / 256, 256, 0, stream>>>(dst, deg);

    // ---- SAGE layer 1: h1 = gelu(mean_aggr(x) @ Wl1 + x @ Wr1 + bl1) ------
    hipMemsetAsync(agg, 0, (size_t)N_NODES * IN_DIM * sizeof(float), stream);
    scatter_kernel<<<N_EDGES, IN_DIM, 0, stream>>>(x, src, dst, agg, IN_DIM);
    fused_gemm<IN_DIM, HIDDEN, 8><<<gGemm, 256, 0, stream>>>(
        agg, Wl1, x, Wr1, bl1, deg, h1, 1);

    // ---- SAGE layer 2: h2 = gelu(mean_aggr(h1) @ Wl2 + h1 @ Wr2 + bl2) ----
    hipMemsetAsync(agg, 0, (size_t)N_NODES * HIDDEN * sizeof(float), stream);
    scatter_kernel<<<N_EDGES, HIDDEN, 0, stream>>>(h1, src, dst, agg, HIDDEN);
    fused_gemm<HIDDEN, HIDDEN, 8><<<gGemm, 256, 0, stream>>>(
        agg, Wl2, h1, Wr2, bl2, deg, h2, 1);

    // ---- classifier Linear+GELU: h1 (reused) = gelu(h2 @ Wc1 + bc1) -------
    fused_gemm<HIDDEN, HIDDEN, 8><<<gGemm, 256, 0, stream>>>(
        h2, Wc1, nullptr, nullptr, bc1, nullptr, h1, 1);

    // ---- final Linear + softmax ------------------------------------------
    mlp2_softmax<<<N_NODES / 16, 128, 0, stream>>>(h1, Wc2, bc2, outp);
}